// Model_Train_74947179316230
// MI455X (gfx1250) — compile-verified
//
#include <hip/hip_runtime.h>
#include <math.h>

// ---------------------------------------------------------------------------
// GraphConv forward on MI455X (gfx1250, wave32).
// - Dense GEMMs: V_WMMA_F32_16X16X4_F32; B staged into LDS with
//   GLOBAL_LOAD_ASYNC_TO_LDS_B32 (ASYNCcnt) in transposed padded layout so the
//   inner loop reads fragments as ds_load_2addr_b64; A streamed from global.
// - Wave-uniform control scalarized via readfirstlane (s_cbranch, no exec mask).
// - BatchNorm column stats fused into the layer-GEMM epilogue; edge scatter via
//   float4 gathers + global_atomic_add_f32 (working set lives in the 192MB L2)
//   with global_prefetch on the edge index stream.
// ---------------------------------------------------------------------------

typedef float v2f __attribute__((ext_vector_type(2)));
typedef float v8f __attribute__((ext_vector_type(8)));

#define D        64      // node dim
#define IN_DIM   128     // input feature dim
#define NCLS     40      // classes
#define NPAD     48      // classes padded to 16-col tiles
#define LEAKY    0.2f
#define BN_EPS   1e-5f

// Async copy of one dword: copies the global dword at gaddr to the LDS address in ldsOff; tracked by ASYNCcnt.
__device__ __forceinline__ void async_g2lds_b32(unsigned ldsOff, const float* gaddr) {
  asm volatile("global_load_async_to_lds_b32 %0, %1, off"
               :: "v"(ldsOff), "v"((unsigned long long)(size_t)gaddr)
               : "memory");
}
__device__ __forceinline__ void async_wait0() {
  asm volatile("s_wait_asynccnt 0x0" ::: "memory");
}

// ---------------------------------------------------------------- utilities
__global__ void zero4_kernel(float4* __restrict__ p, long long n4) {
  long long i = (long long)blockIdx.x * blockDim.x + threadIdx.x;
  const long long stride = (long long)gridDim.x * blockDim.x;
  const float4 z = make_float4(0.f, 0.f, 0.f, 0.f);
  for (; i < n4; i += stride) p[i] = z;
}

__global__ void degree_kernel(const int* __restrict__ src, const int* __restrict__ dst,
                              float* __restrict__ deg_out, float* __restrict__ deg_in, int E) {
  int t = blockIdx.x * blockDim.x + threadIdx.x;
  if (t >= E) return;
  atomicAdd(&deg_out[src[t]], 1.0f);
  atomicAdd(&deg_in[dst[t]], 1.0f);
}

__global__ void norm_kernel(float* __restrict__ deg_out, float* __restrict__ deg_in, int N) {
  int i = blockIdx.x * blockDim.x + threadIdx.x;
  if (i >= N) return;
  float a = deg_out[i];
  float b = deg_in[i];
  deg_out[i] = rsqrtf(a < 1.0f ? 1.0f : a);
  deg_in[i]  = rsqrtf(b < 1.0f ? 1.0f : b);
}

// ------------------------------------------------- WMMA GEMM: h = V@W_in + b
__global__ void gemm_in_wmma(const float* __restrict__ A,   // [N, 128]
                             const float* __restrict__ W,   // [128, 64]
                             const float* __restrict__ bias,// [64]
                             float* __restrict__ out,       // [N, 64]
                             int nstrips, int N) {
  constexpr int K  = IN_DIM;
  constexpr int KP = K + 2;                      // padded stride (bank spread)
  __shared__ float sW[D * KP];                   // sW[n*KP + k] = W[k][n]
  for (int i = threadIdx.x; i < K * D; i += blockDim.x) {
    const int k = i / D, n = i % D;
    async_g2lds_b32((unsigned)((size_t)&sW[n * KP + k]), &W[i]);  // VGPR-free stage
  }
  async_wait0();
  __syncthreads();

  const int lane = threadIdx.x & 31;
  const int strip = __builtin_amdgcn_readfirstlane(
      blockIdx.x * (blockDim.x >> 5) + (threadIdx.x >> 5));  // force SGPR
  if (strip >= nstrips) return;                  // scalar branch

  const int rlo  = lane & 15;
  const int ksel = lane >> 4;                    // 0: K+{0,1}, 1: K+{2,3}
  int arow = strip * 16 + rlo;
  if (arow >= N) arow = N - 1;                   // clamp (load only)
  const float* ap = A + (long long)arow * K;

  v8f acc[4];
  const v8f vzero = {0.f, 0.f, 0.f, 0.f, 0.f, 0.f, 0.f, 0.f};
#pragma unroll
  for (int nt = 0; nt < 4; ++nt) acc[nt] = vzero;

  for (int k0 = 0; k0 < K; k0 += 4) {
    const v2f a = *(const v2f*)&ap[k0 + ksel * 2];              // global_load_b64
#pragma unroll
    for (int nt = 0; nt < 4; ++nt) {
      const int col = nt * 16 + rlo;
      const v2f b = *(const v2f*)&sW[col * KP + k0 + ksel * 2]; // ds_load(2addr)_b64
      acc[nt] = __builtin_amdgcn_wmma_f32_16x16x4_f32(
          false, a, false, b, (short)0, acc[nt], false, false);
    }
  }

  const int rbase = strip * 16 + ksel * 8;       // C layout: M = vgpr + ksel*8
  if (strip * 16 + 16 <= N) {                    // scalar fast path
#pragma unroll
    for (int nt = 0; nt < 4; ++nt) {
      const int col = nt * 16 + rlo;
      const float bv = bias[col];
#pragma unroll
      for (int v = 0; v < 8; ++v)
        out[(long long)(rbase + v) * D + col] = acc[nt][v] + bv;
    }
  } else {
#pragma unroll
    for (int nt = 0; nt < 4; ++nt) {
      const int col = nt * 16 + rlo;
      const float bv = bias[col];
#pragma unroll
      for (int v = 0; v < 8; ++v)
        if (rbase + v < N) out[(long long)(rbase + v) * D + col] = acc[nt][v] + bv;
    }
  }
}

// --------------- WMMA GEMM: hc = (agg*norm_dst)@W + b, fused BN column stats
__global__ void gemm_layer_wmma(const float* __restrict__ agg,     // [N, 64]
                                const float* __restrict__ norm_dst,// [N]
                                const float* __restrict__ W,       // [64, 64]
                                const float* __restrict__ bias,    // [64]
                                float* __restrict__ out,           // [N, 64]
                                float* __restrict__ stats,         // [sum64|sq64|..]
                                int nstrips, int N) {
  constexpr int K  = D;
  constexpr int KP = K + 2;
  __shared__ float sW[D * KP];
  for (int i = threadIdx.x; i < K * D; i += blockDim.x) {
    const int k = i / D, n = i % D;
    async_g2lds_b32((unsigned)((size_t)&sW[n * KP + k]), &W[i]);
  }
  async_wait0();
  __syncthreads();

  const int lane = threadIdx.x & 31;
  const int strip = __builtin_amdgcn_readfirstlane(
      blockIdx.x * (blockDim.x >> 5) + (threadIdx.x >> 5));
  if (strip >= nstrips) return;

  const int rlo  = lane & 15;
  const int ksel = lane >> 4;
  int arow = strip * 16 + rlo;
  if (arow >= N) arow = N - 1;
  const float* ap = agg + (long long)arow * K;
  const float nrm = norm_dst[arow];              // fused D^{-1/2} (dst side)

  v8f acc[4];
  const v8f vzero = {0.f, 0.f, 0.f, 0.f, 0.f, 0.f, 0.f, 0.f};
#pragma unroll
  for (int nt = 0; nt < 4; ++nt) acc[nt] = vzero;

  for (int k0 = 0; k0 < K; k0 += 4) {
    v2f a = *(const v2f*)&ap[k0 + ksel * 2];
    a.x *= nrm; a.y *= nrm;
#pragma unroll
    for (int nt = 0; nt < 4; ++nt) {
      const int col = nt * 16 + rlo;
      const v2f b = *(const v2f*)&sW[col * KP + k0 + ksel * 2];
      acc[nt] = __builtin_amdgcn_wmma_f32_16x16x4_f32(
          false, a, false, b, (short)0, acc[nt], false, false);
    }
  }

  const int rbase = strip * 16 + ksel * 8;
  if (strip * 16 + 16 <= N) {
#pragma unroll
    for (int nt = 0; nt < 4; ++nt) {
      const int col = nt * 16 + rlo;
      const float bv = bias[col];
      float lsum = 0.0f, lsq = 0.0f;
#pragma unroll
      for (int v = 0; v < 8; ++v) {
        const float val = acc[nt][v] + bv;
        out[(long long)(rbase + v) * D + col] = val;
        lsum += val; lsq += val * val;
      }
      atomicAdd(&stats[col], lsum);              // fused BN reduction
      atomicAdd(&stats[64 + col], lsq);
    }
  } else {
#pragma unroll
    for (int nt = 0; nt < 4; ++nt) {
      const int col = nt * 16 + rlo;
      const float bv = bias[col];
      float lsum = 0.0f, lsq = 0.0f;
#pragma unroll
      for (int v = 0; v < 8; ++v) {
        if (rbase + v < N) {
          const float val = acc[nt][v] + bv;
          out[(long long)(rbase + v) * D + col] = val;
          lsum += val; lsq += val * val;
        }
      }
      atomicAdd(&stats[col], lsum);
      atomicAdd(&stats[64 + col], lsq);
    }
  }
}

// ------------------------------ WMMA GEMM: logits = h@W_out + b_out (N=40)
__global__ void gemm_out_wmma(const float* __restrict__ A,    // [N, 64]
                              const float* __restrict__ W,    // [64, 40]
                              const float* __restrict__ bias, // [40]
                              float* __restrict__ out,        // [N, 40]
                              int nstrips, int N) {
  constexpr int K  = D;
  constexpr int KP = K + 2;
  __shared__ float sW[NPAD * KP];                // zero-padded cols 40..47
  for (int i = threadIdx.x; i < K * NPAD; i += blockDim.x) {
    const int k = i / NPAD, n = i % NPAD;
    if (n < NCLS) async_g2lds_b32((unsigned)((size_t)&sW[n * KP + k]), &W[k * NCLS + n]);
    else          sW[n * KP + k] = 0.0f;
  }
  async_wait0();
  __syncthreads();

  const int lane = threadIdx.x & 31;
  const int strip = __builtin_amdgcn_readfirstlane(
      blockIdx.x * (blockDim.x >> 5) + (threadIdx.x >> 5));
  if (strip >= nstrips) return;

  const int rlo  = lane & 15;
  const int ksel = lane >> 4;
  int arow = strip * 16 + rlo;
  if (arow >= N) arow = N - 1;
  const float* ap = A + (long long)arow * K;

  v8f acc[3];
  const v8f vzero = {0.f, 0.f, 0.f, 0.f, 0.f, 0.f, 0.f, 0.f};
#pragma unroll
  for (int nt = 0; nt < 3; ++nt) acc[nt] = vzero;

  for (int k0 = 0; k0 < K; k0 += 4) {
    const v2f a = *(const v2f*)&ap[k0 + ksel * 2];
#pragma unroll
    for (int nt = 0; nt < 3; ++nt) {
      const int col = nt * 16 + rlo;
      const v2f b = *(const v2f*)&sW[col * KP + k0 + ksel * 2];
      acc[nt] = __builtin_amdgcn_wmma_f32_16x16x4_f32(
          false, a, false, b, (short)0, acc[nt], false, false);
    }
  }

  const int rbase = strip * 16 + ksel * 8;
  if (strip * 16 + 16 <= N) {
#pragma unroll
    for (int nt = 0; nt < 3; ++nt) {
      const int col = nt * 16 + rlo;
      if (col < NCLS) {                          // lane-divergent only on tile 2
        const float bv = bias[col];
#pragma unroll
        for (int v = 0; v < 8; ++v)
          out[(long long)(rbase + v) * NCLS + col] = acc[nt][v] + bv;
      }
    }
  } else {
#pragma unroll
    for (int nt = 0; nt < 3; ++nt) {
      const int col = nt * 16 + rlo;
      if (col < NCLS) {
        const float bv = bias[col];
#pragma unroll
        for (int v = 0; v < 8; ++v)
          if (rbase + v < N) out[(long long)(rbase + v) * NCLS + col] = acc[nt][v] + bv;
      }
    }
  }
}

// --------------------------- edge scatter: agg[dst] += h[src] * norm_src[src]
__global__ void scatter_kernel(const int* __restrict__ src, const int* __restrict__ dst,
                               const float* __restrict__ h, const float* __restrict__ norm_src,
                               float* __restrict__ agg, long long total /* E*16 */,
                               int E) {
  long long t = (long long)blockIdx.x * blockDim.x + threadIdx.x;
  if (t >= total) return;
  const int e = (int)(t >> 4);
  const int p = (int)(t & 15);
  if (p == 0 && e + 4096 < E) {                  // stream edge indices ahead
    __builtin_prefetch(&src[e + 4096], 0, 0);
    __builtin_prefetch(&dst[e + 4096], 0, 0);
  }
  const int s = src[e];
  const int d = dst[e];
  const float ns = norm_src[s];
  const float4 v = *reinterpret_cast<const float4*>(h + (long long)s * D + p * 4);
  float* ap = agg + (long long)d * D + p * 4;
  atomicAdd(ap + 0, v.x * ns);
  atomicAdd(ap + 1, v.y * ns);
  atomicAdd(ap + 2, v.z * ns);
  atomicAdd(ap + 3, v.w * ns);
}

__global__ void finalize_stats(float* __restrict__ stats, float invN) {
  const int c = threadIdx.x;   // 64 threads
  const float mean = stats[c] * invN;
  const float var  = stats[64 + c] * invN - mean * mean;  // biased, as reference
  stats[128 + c] = mean;
  stats[192 + c] = rsqrtf(var + BN_EPS);
}

// --------------------- h = leakyrelu(batchnorm(hc)) + h   (residual, in place)
// float4 per thread: 4 consecutive columns share the row.
__global__ void bn_act_res_kernel(const float4* __restrict__ hc, const float* __restrict__ stats,
                                  const float* __restrict__ gamma, const float* __restrict__ beta,
                                  float4* __restrict__ h, long long total4) {
  long long i = (long long)blockIdx.x * blockDim.x + threadIdx.x;
  const long long stride = (long long)gridDim.x * blockDim.x;
  for (; i < total4; i += stride) {
    const int c = (int)((i * 4) & (D - 1));
    const float4 x = hc[i];
    float4 r = h[i];
    float v0 = (x.x - stats[128 + c + 0]) * stats[192 + c + 0] * gamma[c + 0] + beta[c + 0];
    float v1 = (x.y - stats[128 + c + 1]) * stats[192 + c + 1] * gamma[c + 1] + beta[c + 1];
    float v2 = (x.z - stats[128 + c + 2]) * stats[192 + c + 2] * gamma[c + 2] + beta[c + 2];
    float v3 = (x.w - stats[128 + c + 3]) * stats[192 + c + 3] * gamma[c + 3] + beta[c + 3];
    v0 = (v0 >= 0.f) ? v0 : LEAKY * v0;
    v1 = (v1 >= 0.f) ? v1 : LEAKY * v1;
    v2 = (v2 >= 0.f) ? v2 : LEAKY * v2;
    v3 = (v3 >= 0.f) ? v3 : LEAKY * v3;
    r.x += v0; r.y += v1; r.z += v2; r.w += v3;
    h[i] = r;
  }
}

// ----------------------------------------------------- per-row log-softmax
__global__ void logsoftmax_kernel(const float* __restrict__ logits,
                                  float* __restrict__ out, int N) {
  const int r = blockIdx.x * blockDim.x + threadIdx.x;
  if (r >= N) return;
  float x[NCLS];
  float mx = -3.0e38f;
  const float* row = logits + (long long)r * NCLS;
#pragma unroll
  for (int c = 0; c < NCLS; ++c) { x[c] = row[c]; mx = fmaxf(mx, x[c]); }
  float s = 0.0f;
#pragma unroll
  for (int c = 0; c < NCLS; ++c) s += __expf(x[c] - mx);
  const float lse = mx + __logf(s);
  float* o = out + (long long)r * NCLS;
#pragma unroll
  for (int c = 0; c < NCLS; ++c) o[c] = x[c] - lse;
}

// ===========================================================================
extern "C" void kernel_launch(void* const* d_in, const int* in_sizes, int n_in,
                              void* d_out, int out_size, void* d_ws, size_t ws_size,
                              hipStream_t stream) {
  const float* V     = (const float*)d_in[0];
  const int*   src   = (const int*)  d_in[1];
  const int*   dst   = (const int*)  d_in[2];
  const float* W_in  = (const float*)d_in[3];
  const float* b_in  = (const float*)d_in[4];
  const float* W_l   = (const float*)d_in[5];
  const float* b_l   = (const float*)d_in[6];
  const float* gamma = (const float*)d_in[7];
  const float* beta  = (const float*)d_in[8];
  const float* W_out = (const float*)d_in[9];
  const float* b_out = (const float*)d_in[10];
  float* out = (float*)d_out;

  const int N = in_sizes[0] / IN_DIM;   // 100000
  const int E = in_sizes[1];            // 1600000
  const int N_LAYERS = in_sizes[6] / D; // 4

  // ---- workspace carve-out (256B aligned)
  char* ws = (char*)d_ws;
  size_t off = 0;
  auto carve = [&](size_t bytes) -> float* {
    float* p = (float*)(ws + off);
    off = (off + bytes + 255) & ~((size_t)255);
    return p;
  };
  float* deg     = carve((size_t)2 * N * sizeof(float)); // deg_out | deg_in (-> norms)
  float* deg_out = deg;
  float* deg_in  = deg + N;
  float* h       = carve((size_t)N * D * sizeof(float));
  float* agg     = carve((size_t)N * D * sizeof(float));
  float* hc      = carve((size_t)N * D * sizeof(float));
  float* stats   = carve(256 * sizeof(float));           // sum|sumsq|mean|invstd
  float* logits  = carve((size_t)N * NCLS * sizeof(float));

  const long long totalND = (long long)N * D;
  const int nstrips = (N + 15) / 16;
  const int gemmGrid = (nstrips + 7) / 8;   // 8 waves (256 thr) per block

  // ---- degrees & norms
  zero4_kernel<<<512, 256, 0, stream>>>((float4*)deg, (long long)2 * N / 4);
  degree_kernel<<<(E + 255) / 256, 256, 0, stream>>>(src, dst, deg_out, deg_in, E);
  norm_kernel<<<(N + 255) / 256, 256, 0, stream>>>(deg_out, deg_in, N);

  // ---- input embedding (WMMA)
  gemm_in_wmma<<<gemmGrid, 256, 0, stream>>>(V, W_in, b_in, h, nstrips, N);

  // ---- GraphConv layers
  const long long edgeWork = (long long)E * 16;
  const int scatterGrid = (int)((edgeWork + 255) / 256);
  for (int l = 0; l < N_LAYERS; ++l) {
    zero4_kernel<<<1024, 256, 0, stream>>>((float4*)agg, totalND / 4);
    scatter_kernel<<<scatterGrid, 256, 0, stream>>>(src, dst, h, deg_out, agg, edgeWork, E);
    zero4_kernel<<<1, 64, 0, stream>>>((float4*)stats, 64);
    gemm_layer_wmma<<<gemmGrid, 256, 0, stream>>>(agg, deg_in, W_l + (size_t)l * D * D,
                                                  b_l + (size_t)l * D, hc, stats, nstrips, N);
    finalize_stats<<<1, 64, 0, stream>>>(stats, 1.0f / (float)N);
    bn_act_res_kernel<<<2048, 256, 0, stream>>>((const float4*)hc, stats, gamma + (size_t)l * D,
                                                beta + (size_t)l * D, (float4*)h, totalND / 4);
  }

  // ---- classifier head (WMMA) + log-softmax
  gemm_out_wmma<<<gemmGrid, 256, 0, stream>>>(h, W_out, b_out, logits, nstrips, N);
  logsoftmax_kernel<<<(N + 127) / 128, 128, 0, stream>>>(logits, out, N);
}